// MaskedMSELoss_3710851744149
// MI455X (gfx1250) — compile-verified
//
#include <hip/hip_runtime.h>
#include <hip/hip_bf16.h>

typedef __attribute__((ext_vector_type(2))) float v2f;
typedef __attribute__((ext_vector_type(8))) float v8f;

#define NBLK 1024
#define NTHR 256

// Pass 1: streaming masked-SSE partial reduction. 128-bit loads, grid-stride,
// wave32 shuffle tree -> LDS -> one (sum2, cnt) pair per block. Deterministic.
__global__ __launch_bounds__(NTHR) void masked_mse_partial(
    const float* __restrict__ in, const float* __restrict__ tgt,
    const int* __restrict__ mask, float* __restrict__ psum,
    float* __restrict__ pcnt, int n)
{
    const int tid    = blockIdx.x * blockDim.x + threadIdx.x;
    const int stride = gridDim.x * blockDim.x;
    const int n4     = n >> 2;

    const float4* __restrict__ in4 = (const float4*)in;
    const float4* __restrict__ tg4 = (const float4*)tgt;
    const int4*   __restrict__ mk4 = (const int4*)mask;

    float s = 0.0f, c = 0.0f;

    for (int i = tid; i < n4; i += stride) {
        float4 a = in4[i];
        float4 b = tg4[i];
        int4   m = mk4[i];
        float d0 = a.x - b.x, d1 = a.y - b.y, d2 = a.z - b.z, d3 = a.w - b.w;
        float m0 = (m.x == 1) ? 1.0f : 0.0f;
        float m1 = (m.y == 1) ? 1.0f : 0.0f;
        float m2 = (m.z == 1) ? 1.0f : 0.0f;
        float m3 = (m.w == 1) ? 1.0f : 0.0f;
        s = fmaf(d0 * d0, m0, s);
        s = fmaf(d1 * d1, m1, s);
        s = fmaf(d2 * d2, m2, s);
        s = fmaf(d3 * d3, m3, s);
        c += m0 + m1 + m2 + m3;
    }
    // scalar tail (n not divisible by 4) — grid-stride, deterministic
    for (int i = (n4 << 2) + tid; i < n; i += stride) {
        float d = in[i] - tgt[i];
        float m = (mask[i] == 1) ? 1.0f : 0.0f;
        s = fmaf(d * d, m, s);
        c += m;
    }

    // wave32 reduction
    #pragma unroll
    for (int off = 16; off > 0; off >>= 1) {
        s += __shfl_down(s, off, 32);
        c += __shfl_down(c, off, 32);
    }

    __shared__ float ls[NTHR / 32];
    __shared__ float lc[NTHR / 32];
    const int lane = threadIdx.x & 31;
    const int wave = threadIdx.x >> 5;
    if (lane == 0) { ls[wave] = s; lc[wave] = c; }
    __syncthreads();

    if (wave == 0) {
        s = (lane < (NTHR / 32)) ? ls[lane] : 0.0f;
        c = (lane < (NTHR / 32)) ? lc[lane] : 0.0f;
        #pragma unroll
        for (int off = (NTHR / 64); off > 0; off >>= 1) {
            s += __shfl_down(s, off, 32);
            c += __shfl_down(c, off, 32);
        }
        if (lane == 0) {
            psum[blockIdx.x] = s;
            pcnt[blockIdx.x] = c;
        }
    }
}

// Pass 2: one wave32. WMMA f32 16x16x4 with B == ones acts as a full-precision
// accumulating reducer: D = A*1 + C, each row-sum replicated across columns.
// 64 partials per WMMA; C chains across iterations. Sum of ALL entries of A is
// what we want, so the lane<->element bijection is irrelevant with B = ones.
// EXEC all-ones: 32 threads, no divergence before the WMMAs.
__global__ __launch_bounds__(32) void masked_mse_final(
    const float* __restrict__ psum, const float* __restrict__ pcnt,
    float* __restrict__ out, int nblk)
{
    const int lane = threadIdx.x; // 0..31

    v8f accS = {};
    v8f accC = {};
    v2f ones;
    ones[0] = 1.0f;
    ones[1] = 1.0f;

    // nblk == NBLK == 1024, multiple of 64 -> no bounds checks, EXEC stays full
    for (int base = 0; base < nblk; base += 64) {
        v2f aS, aC;
        aS[0] = psum[base + lane];
        aS[1] = psum[base + 32 + lane];
        aC[0] = pcnt[base + lane];
        aC[1] = pcnt[base + 32 + lane];
        // 8 args: (neg_a, A, neg_b, B, c_mod, C, reuse_a, reuse_b)
        accS = __builtin_amdgcn_wmma_f32_16x16x4_f32(
            false, aS, false, ones, (short)0, accS, false, false);
        accC = __builtin_amdgcn_wmma_f32_16x16x4_f32(
            false, aC, false, ones, (short)0, accC, false, false);
    }

    // D layout: VGPR r holds rows M=r (lanes 0-15) and M=r+8 (lanes 16-31),
    // each row-sum replicated across columns. Sum 8 VGPRs in-lane, then add
    // the opposite half-wave to get the total of all 16 rows.
    float s = accS[0] + accS[1] + accS[2] + accS[3] +
              accS[4] + accS[5] + accS[6] + accS[7];
    float c = accC[0] + accC[1] + accC[2] + accC[3] +
              accC[4] + accC[5] + accC[6] + accC[7];
    s += __shfl_xor(s, 16, 32);
    c += __shfl_xor(c, 16, 32);

    if (lane == 0) out[0] = s / c;
}

extern "C" void kernel_launch(void* const* d_in, const int* in_sizes, int n_in,
                              void* d_out, int out_size, void* d_ws, size_t ws_size,
                              hipStream_t stream) {
    const float* in   = (const float*)d_in[0];
    const float* tgt  = (const float*)d_in[1];
    const int*   mask = (const int*)d_in[2];
    float*       out  = (float*)d_out;
    const int    n    = in_sizes[0];

    float* psum = (float*)d_ws;        // NBLK floats
    float* pcnt = psum + NBLK;         // NBLK floats (8 KB total)

    masked_mse_partial<<<NBLK, NTHR, 0, stream>>>(in, tgt, mask, psum, pcnt, n);
    masked_mse_final<<<1, 32, 0, stream>>>(psum, pcnt, out, NBLK);
}